// DecoderTP_41077067219680
// MI455X (gfx1250) — compile-verified
//
#include <hip/hip_runtime.h>
#include <hip/hip_bf16.h>

#define EMBED_DIM 256
#define NUM_SURV  20
#define NUM_EVENTS 8192
#define EPSF 1e-7f

typedef __attribute__((ext_vector_type(2))) float v2f;
typedef __attribute__((ext_vector_type(8))) float v8f;

__device__ __forceinline__ float thp_softplus(float g, float ps) {
    float x = g / (ps + EPSF);
    x = fminf(fmaxf(x, -75.0f), 75.0f);
    return ps * log1pf(expf(x));   // psi * softplus(clip(g/(psi+eps)))
}

// Async-stage the raw 2x512 W matrix (4096 B) into LDS via the CDNA5
// async-to-LDS path (ASYNCcnt), then fold into lds_w[k*2+n] = 0.5*(Wu+Wv).
__device__ __forceinline__ void stage_w_lds(const float* __restrict__ W,
                                            float* lds_raw, float* lds_w,
                                            int tid) {
    for (int ofs = tid * 16; ofs < 4096; ofs += 128 * 16) {
        unsigned lds_dst = (unsigned)(uintptr_t)((char*)lds_raw + ofs);
        unsigned long long gsrc = (unsigned long long)(uintptr_t)((const char*)W + ofs);
        asm volatile("global_load_async_to_lds_b128 %0, %1, off"
                     :: "v"(lds_dst), "v"(gsrc) : "memory");
    }
    asm volatile("s_wait_asynccnt 0x0" ::: "memory");
    __syncthreads();
    for (int j = tid; j < 2 * EMBED_DIM; j += 128) {
        const int n = j & 1, k = j >> 1;
        lds_w[j] = 0.5f * (lds_raw[n * 512 + k] + lds_raw[n * 512 + 256 + k]);
    }
    __syncthreads();
}

// ---------------------------------------------------------------------------
// Kernel 1: survival intensities.  Each wave owns 16 events and loops over all
// S=20 survival steps (rows s*E+e are contiguous per step).  Dot products via
// V_WMMA_F32_16X16X4_F32: A = 16 events x 4 K (fp32), B columns 0/1 carry the
// folded weight rows w0/w1, two C accumulators (even/odd chunks) for ILP.
// ---------------------------------------------------------------------------
__global__ __launch_bounds__(128) void thp_surv_kernel(
    const float* __restrict__ u_non, const float* __restrict__ v_non,
    const float* __restrict__ last_update, const float* __restrict__ cur_time,
    const float* __restrict__ W, const float* __restrict__ b,
    const float* __restrict__ psi, const float* __restrict__ alpha,
    const int* __restrict__ assoc, const int* __restrict__ src,
    const int* __restrict__ pos_dst, const int* __restrict__ et,
    float* __restrict__ ws_integral)
{
    __shared__ float lds_raw[2 * 512];         // async-staged raw W
    __shared__ float lds_w[2 * EMBED_DIM];     // lds_w[k*2+n] = 0.5*(W[n][k]+W[n][256+k])
    __shared__ float lds_c[4][256];            // per-wave C staging

    const int tid = threadIdx.x;
    stage_w_lds(W, lds_raw, lds_w, tid);

    const int wave = tid >> 5;
    const int lane = tid & 31;
    const int m    = lane & 15;          // event row within tile; also B column n
    const int off  = (lane >> 4) << 1;   // K sub-offset for this half-wave: 0 or 2

    const int e = blockIdx.x * 64 + wave * 16 + m;

    // Per-event scalars (lanes m and m+16 compute the same values; cheap).
    const int   eb   = (et[e] > 0) ? 1 : 0;
    const int   isrc = assoc[src[e]];
    const int   idst = assoc[pos_dst[e]];
    const float ltp  = fmaxf(last_update[isrc], last_update[idst]);
    const float td   = cur_time[e] - ltp;
    const float tdss = td * (1.0f / (float)(NUM_SURV - 1));
    const float inv1 = 1.0f / (ltp + 1.0f);
    const float bb = b[eb], ps = psi[eb], al = alpha[eb];

    // Loop-invariant B-column select: unconditional LDS reads, VALU mask.
    const int    nsel = (m < 2) ? m : 0;
    const float  bmsk = (m < 2) ? 1.0f : 0.0f;
    const float* wl   = &lds_w[nsel];          // index with k*2

    const size_t rowbase = (size_t)e * EMBED_DIM + (size_t)off;
    float acc = 0.0f;

    for (int s = 0; s < NUM_SURV; ++s) {
        v8f c0 = {}, c1 = {};
        const float* up = u_non + (size_t)s * (NUM_EVENTS * EMBED_DIM) + rowbase;
        const float* vp = v_non + (size_t)s * (NUM_EVENTS * EMBED_DIM) + rowbase;
        #pragma unroll 4
        for (int ck = 0; ck < 64; ck += 2) {
            const int k0 = ck * 4 + off;           // even chunk K base
            const int k1 = k0 + 4;                 // odd chunk K base
            v2f au0 = *(const v2f*)(up + ck * 4);
            v2f av0 = *(const v2f*)(vp + ck * 4);
            v2f au1 = *(const v2f*)(up + ck * 4 + 4);
            v2f av1 = *(const v2f*)(vp + ck * 4 + 4);
            v2f a0 = au0 + av0;
            v2f a1 = au1 + av1;
            v2f bf0, bf1;
            bf0.x = bmsk * wl[(k0)     * 2];
            bf0.y = bmsk * wl[(k0 + 1) * 2];
            bf1.x = bmsk * wl[(k1)     * 2];
            bf1.y = bmsk * wl[(k1 + 1) * 2];
            c0 = __builtin_amdgcn_wmma_f32_16x16x4_f32(false, a0, false, bf0,
                                                       (short)0, c0, false, false);
            c1 = __builtin_amdgcn_wmma_f32_16x16x4_f32(false, a1, false, bf1,
                                                       (short)0, c1, false, false);
        }
        v8f c = c0 + c1;
        #pragma unroll
        for (int j = 0; j < 8; ++j) lds_c[wave][j * 32 + lane] = c[j];
        __syncthreads();
        if (lane < 16) {
            const int base = (m & 7) * 32 + ((m >> 3) << 4);
            const float G0 = lds_c[wave][base + 0];
            const float G1 = lds_c[wave][base + 1];
            const float g  = (eb ? G1 : G0) + bb + al * ((float)s * tdss * inv1);
            acc += thp_softplus(g, ps);
        }
        __syncthreads();
    }
    if (lane < 16) ws_integral[e] = tdss * acc;          // integral per event
}

// ---------------------------------------------------------------------------
// Kernel 2: lambda_uv (rows 0..E-1) and lambda_uv_neg (rows E..2E-1) with
// gathered embedding rows; identical WMMA machinery, single K pass per row.
// ---------------------------------------------------------------------------
__global__ __launch_bounds__(128) void thp_uv_kernel(
    const float* __restrict__ all_emb,
    const float* __restrict__ last_update, const float* __restrict__ cur_time,
    const float* __restrict__ W, const float* __restrict__ b,
    const float* __restrict__ psi, const float* __restrict__ alpha,
    const int* __restrict__ assoc, const int* __restrict__ src,
    const int* __restrict__ pos_dst, const int* __restrict__ neg_dst,
    const int* __restrict__ et,
    float* __restrict__ lam_pos, float* __restrict__ lam_neg)
{
    __shared__ float lds_raw[2 * 512];
    __shared__ float lds_w[2 * EMBED_DIM];
    __shared__ float lds_c[4][256];

    const int tid = threadIdx.x;
    stage_w_lds(W, lds_raw, lds_w, tid);

    const int wave = tid >> 5;
    const int lane = tid & 31;
    const int m    = lane & 15;
    const int off  = (lane >> 4) << 1;

    const int r      = blockIdx.x * 64 + wave * 16 + m;   // 0..2E-1
    const int is_neg = (r >= NUM_EVENTS) ? 1 : 0;
    const int e      = is_neg ? (r - NUM_EVENTS) : r;

    const int   eb   = (et[e] > 0) ? 1 : 0;
    const int   isrc = assoc[src[e]];
    const int   dsel = is_neg ? neg_dst[e] : pos_dst[e];
    const int   idst = assoc[dsel];
    const float ltp  = fmaxf(last_update[isrc], last_update[idst]);
    const float temporal = (cur_time[e] - ltp) / (ltp + 1.0f);
    const float bb = b[eb], ps = psi[eb], al = alpha[eb];

    const int    nsel = (m < 2) ? m : 0;
    const float  bmsk = (m < 2) ? 1.0f : 0.0f;
    const float* wl   = &lds_w[nsel];

    const float* up = all_emb + (size_t)isrc * EMBED_DIM + (size_t)off;
    const float* vp = all_emb + (size_t)idst * EMBED_DIM + (size_t)off;

    v8f c0 = {}, c1 = {};
    #pragma unroll 4
    for (int ck = 0; ck < 64; ck += 2) {
        const int k0 = ck * 4 + off;
        const int k1 = k0 + 4;
        v2f au0 = *(const v2f*)(up + ck * 4);
        v2f av0 = *(const v2f*)(vp + ck * 4);
        v2f au1 = *(const v2f*)(up + ck * 4 + 4);
        v2f av1 = *(const v2f*)(vp + ck * 4 + 4);
        v2f a0 = au0 + av0;
        v2f a1 = au1 + av1;
        v2f bf0, bf1;
        bf0.x = bmsk * wl[(k0)     * 2];
        bf0.y = bmsk * wl[(k0 + 1) * 2];
        bf1.x = bmsk * wl[(k1)     * 2];
        bf1.y = bmsk * wl[(k1 + 1) * 2];
        c0 = __builtin_amdgcn_wmma_f32_16x16x4_f32(false, a0, false, bf0,
                                                   (short)0, c0, false, false);
        c1 = __builtin_amdgcn_wmma_f32_16x16x4_f32(false, a1, false, bf1,
                                                   (short)0, c1, false, false);
    }
    v8f c = c0 + c1;
    #pragma unroll
    for (int j = 0; j < 8; ++j) lds_c[wave][j * 32 + lane] = c[j];
    __syncthreads();
    if (lane < 16) {
        const int base = (m & 7) * 32 + ((m >> 3) << 4);
        const float G0 = lds_c[wave][base + 0];
        const float G1 = lds_c[wave][base + 1];
        const float g  = (eb ? G1 : G0) + bb + al * temporal;
        const float lam = thp_softplus(g, ps);
        if (is_neg) lam_neg[e] = lam; else lam_pos[e] = lam;
    }
}

// ---------------------------------------------------------------------------
// Kernel 3: per-event epilogue + fixed-shape block reductions for the losses.
// ---------------------------------------------------------------------------
__global__ __launch_bounds__(256) void thp_final_kernel(
    const float* __restrict__ ws_integral,
    const float* __restrict__ lam_pos, const float* __restrict__ lam_neg,
    float* __restrict__ out, float* __restrict__ partials)
{
    __shared__ float red_ll[256];
    __shared__ float red_ls[256];
    const int tid = threadIdx.x;
    const int e   = blockIdx.x * 256 + tid;

    const float integ = ws_integral[e];
    const float sv    = expf(-integ);
    const float lp    = lam_pos[e];
    const float ln    = lam_neg[e];
    out[2 + e]              = lp * sv;   // cond_pos
    out[2 + NUM_EVENTS + e] = ln * sv;   // cond_neg

    red_ll[tid] = -logf(lp + EPSF);
    red_ls[tid] = integ;
    __syncthreads();
    #pragma unroll
    for (int st = 128; st > 0; st >>= 1) {
        if (tid < st) {
            red_ll[tid] += red_ll[tid + st];
            red_ls[tid] += red_ls[tid + st];
        }
        __syncthreads();
    }
    if (tid == 0) {
        partials[blockIdx.x * 2 + 0] = red_ll[0];
        partials[blockIdx.x * 2 + 1] = red_ls[0];
    }
}

__global__ void thp_loss_kernel(const float* __restrict__ partials,
                                float* __restrict__ out)
{
    if (threadIdx.x == 0 && blockIdx.x == 0) {
        float ll = 0.0f, ls = 0.0f;
        for (int i = 0; i < 32; ++i) {
            ll += partials[2 * i + 0];
            ls += partials[2 * i + 1];
        }
        out[0] = ll / (float)NUM_EVENTS;   // loss_lambda / E
        out[1] = ls / (float)NUM_EVENTS;   // loss_surv   / E
    }
}

extern "C" void kernel_launch(void* const* d_in, const int* in_sizes, int n_in,
                              void* d_out, int out_size, void* d_ws, size_t ws_size,
                              hipStream_t stream) {
    (void)in_sizes; (void)n_in; (void)out_size; (void)ws_size;
    const float* all_emb     = (const float*)d_in[0];
    const float* u_non       = (const float*)d_in[1];
    const float* v_non       = (const float*)d_in[2];
    const float* last_update = (const float*)d_in[3];
    const float* cur_time    = (const float*)d_in[4];
    const float* W           = (const float*)d_in[5];
    const float* b           = (const float*)d_in[6];
    const float* psi         = (const float*)d_in[7];
    const float* alpha       = (const float*)d_in[8];
    const int*   assoc       = (const int*)d_in[9];
    const int*   src         = (const int*)d_in[10];
    const int*   pos_dst     = (const int*)d_in[11];
    const int*   neg_dst     = (const int*)d_in[12];
    const int*   et          = (const int*)d_in[13];

    float* out = (float*)d_out;
    float* wsf = (float*)d_ws;
    float* ws_integral = wsf;                       // E floats
    float* ws_lam_pos  = wsf + NUM_EVENTS;          // E floats
    float* ws_lam_neg  = wsf + 2 * NUM_EVENTS;      // E floats
    float* ws_partials = wsf + 3 * NUM_EVENTS;      // 64 floats

    // 8192 events / 64 per block
    thp_surv_kernel<<<NUM_EVENTS / 64, 128, 0, stream>>>(
        u_non, v_non, last_update, cur_time, W, b, psi, alpha,
        assoc, src, pos_dst, et, ws_integral);

    // 2*E rows (pos then neg) / 64 per block
    thp_uv_kernel<<<(2 * NUM_EVENTS) / 64, 128, 0, stream>>>(
        all_emb, last_update, cur_time, W, b, psi, alpha,
        assoc, src, pos_dst, neg_dst, et, ws_lam_pos, ws_lam_neg);

    thp_final_kernel<<<NUM_EVENTS / 256, 256, 0, stream>>>(
        ws_integral, ws_lam_pos, ws_lam_neg, out, ws_partials);

    thp_loss_kernel<<<1, 32, 0, stream>>>(ws_partials, out);
}